// AdditiveAttention_31619549233722
// MI455X (gfx1250) — compile-verified
//
#include <hip/hip_runtime.h>
#include <hip/hip_bf16.h>

// Problem constants (match reference setup_inputs)
#define BATCH 4
#define QLEN  1024
#define KLEN  1024
#define DIN   256
#define NH    64
#define DV    256

typedef float v2f __attribute__((ext_vector_type(2)));
typedef float v8f __attribute__((ext_vector_type(8)));

// ---------------------------------------------------------------------------
// fast tanh: prefer gfx1250 v_tanh_f32 if the builtin exists, else
// tanh(x) = 1 - 2/(1+e^{2x}) via v_exp_f32 + v_rcp_f32 (saturates correctly).
// ---------------------------------------------------------------------------
__device__ __forceinline__ float fast_tanh(float x) {
#if __has_builtin(__builtin_amdgcn_tanhf)
    return __builtin_amdgcn_tanhf(x);
#else
    // e^{2x} = exp2(2*log2(e) * x)
    float e = __builtin_amdgcn_exp2f(x * 2.885390081777927f);
    return 1.0f - 2.0f * __builtin_amdgcn_rcpf(e + 1.0f);
#endif
}

// ---------------------------------------------------------------------------
// Kernel 1: projection  out[r,h] = sum_d X[r,d] * W[d,h]
// block = 256 threads = 4 rows x 64 h ; grid.x = rows/4
// ---------------------------------------------------------------------------
__global__ void proj_kernel(const float* __restrict__ X,
                            const float* __restrict__ W,
                            float* __restrict__ out) {
    __shared__ float xrow[4][DIN];
    const int t  = threadIdx.x;
    const int rb = blockIdx.x * 4;

    for (int i = t; i < 4 * DIN; i += 256) {
        int r = i >> 8, c = i & (DIN - 1);
        xrow[r][c] = X[(size_t)(rb + r) * DIN + c];
    }
    __syncthreads();

    const int h = t & (NH - 1);
    const int r = t >> 6;
    float acc = 0.0f;
#pragma unroll 8
    for (int d = 0; d < DIN; ++d)
        acc = fmaf(xrow[r][d], W[(size_t)d * NH + h], acc);
    out[(size_t)(rb + r) * NH + h] = acc;
}

// ---------------------------------------------------------------------------
// Kernel 2: scores[b,q,k] = sum_h wv[h] * tanh(qproj[b,q,h] + kproj[b,k,h])
// block = 256 threads = 16x16 (q,k) tile ; grid = (K/16, Q/16, B)
// LDS rows padded to 65 floats to avoid 16-way bank conflicts on k rows.
// This is the dominant kernel (~268M tanh lanes) -> pure VALU/trans work.
// ---------------------------------------------------------------------------
__global__ void score_kernel(const float* __restrict__ qproj,
                             const float* __restrict__ kproj,
                             const float* __restrict__ wv,
                             float* __restrict__ scores) {
    __shared__ float qs[16][NH + 1];
    __shared__ float ks[16][NH + 1];
    __shared__ float wvs[NH];

    const int t     = threadIdx.x;
    const int b     = blockIdx.z;
    const int qbase = blockIdx.y * 16;
    const int kbase = blockIdx.x * 16;

    for (int i = t; i < 16 * NH; i += 256) {
        int r = i >> 6, c = i & (NH - 1);
        qs[r][c] = qproj[((size_t)(b * QLEN + qbase + r)) * NH + c];
        ks[r][c] = kproj[((size_t)(b * KLEN + kbase + r)) * NH + c];
    }
    if (t < NH) wvs[t] = wv[t];
    __syncthreads();

    const int tx = t & 15;   // k within tile
    const int ty = t >> 4;   // q within tile

    float s = 0.0f;
#pragma unroll 16
    for (int h = 0; h < NH; ++h) {
        float x = qs[ty][h] + ks[tx][h];
        s = fmaf(wvs[h], fast_tanh(x), s);
    }
    scores[((size_t)(b * QLEN + qbase + ty)) * KLEN + kbase + tx] = s;
}

// ---------------------------------------------------------------------------
// Kernel 3: masked softmax over K (in place).
// block = 256 threads per (q,b) row ; grid = (Q, B). 4 elements per thread.
// ---------------------------------------------------------------------------
__global__ void softmax_kernel(float* __restrict__ scores,
                               const int* __restrict__ valid_lens) {
    __shared__ float red[256];
    const int t  = threadIdx.x;
    const int q  = blockIdx.x;
    const int b  = blockIdx.y;
    const int vl = valid_lens[b];

    float* row = scores + ((size_t)(b * QLEN + q)) * KLEN;

    float v[4];
    float m = -3.4e38f;
#pragma unroll
    for (int i = 0; i < 4; ++i) {
        int pos = t + i * 256;
        float x = row[pos];
        if (pos >= vl) x = -1.0e6f;   // reference NEG mask value
        v[i] = x;
        m = fmaxf(m, x);
    }
    red[t] = m;
    __syncthreads();
    for (int s = 128; s > 0; s >>= 1) {
        if (t < s) red[t] = fmaxf(red[t], red[t + s]);
        __syncthreads();
    }
    m = red[0];
    __syncthreads();

    float sum = 0.0f;
#pragma unroll
    for (int i = 0; i < 4; ++i) {
        v[i] = __builtin_amdgcn_exp2f((v[i] - m) * 1.4426950408889634f);
        sum += v[i];
    }
    red[t] = sum;
    __syncthreads();
    for (int s = 128; s > 0; s >>= 1) {
        if (t < s) red[t] += red[t + s];
        __syncthreads();
    }
    const float inv = 1.0f / red[0];
#pragma unroll
    for (int i = 0; i < 4; ++i)
        row[t + i * 256] = v[i] * inv;
}

// ---------------------------------------------------------------------------
// Kernel 4: out[b,q,v] = sum_k attn[b,q,k] * values[b,k,v]
// One wave32 computes a 16x32 output tile: two independent 16x16 accumulator
// chains (c0,c1) share every A fragment -> halves A traffic, 2x wmma density,
// and two independent V_WMMA_F32_16X16X4_F32 dependency chains per wave.
//   A 16x4 f32 layout (ISA 7.12.2): lanes 0-15 M=lane, VGPR0=K0 VGPR1=K1;
//                                   lanes 16-31 M=lane-16, VGPR0=K2 VGPR1=K3.
//   B 4x16 f32: rows striped across lanes: VGPR0 = rows {0,2}, VGPR1 = {1,3}.
//   D 16x16 f32: VGPR r -> M=r (lanes 0-15) / M=r+8 (lanes 16-31), N=lane%16.
// block = 128 (4 waves, 4 q-tiles) ; grid = (DV/32, Q/64, B)
// ---------------------------------------------------------------------------
__global__ void av_wmma_kernel(const float* __restrict__ attn,
                               const float* __restrict__ values,
                               float* __restrict__ out) {
    const int lane  = threadIdx.x & 31;
    const int wave  = threadIdx.x >> 5;
    const int b     = blockIdx.z;
    const int vbase = blockIdx.x * 32;
    const int qbase = (blockIdx.y * 4 + wave) * 16;

    const int half = lane >> 4;   // 0: K pair {0,1}; 1: K pair {2,3}
    const int lrow = lane & 15;   // M for A, N for B/D

    const float* arow  = attn   + ((size_t)(b * QLEN + qbase + lrow)) * KLEN + 2 * half;
    const float* bcol0 = values + ((size_t)(b * KLEN + 2 * half)) * DV + vbase + lrow;
    const float* bcol1 = bcol0 + 16;

    v8f c0 = {0.f, 0.f, 0.f, 0.f, 0.f, 0.f, 0.f, 0.f};
    v8f c1 = {0.f, 0.f, 0.f, 0.f, 0.f, 0.f, 0.f, 0.f};

    // outer strips of 256 K-values; prefetch the next strip of this lane's
    // attn row (uniform guarded branch -> no speculative OOB read)
    for (int kb = 0; kb < KLEN; kb += 256) {
        if (kb + 256 < KLEN)
            __builtin_prefetch(arow + kb + 256, 0, 1);   // global_prefetch_b8

#pragma unroll 4
        for (int kk = kb; kk < kb + 256; kk += 4) {
            // A: two consecutive f32 of this lane's attn row -> one b64 load
            v2f a = *(const v2f*)(arow + kk);
            // B: two rows of values, 2x16 contiguous floats across lanes
            v2f b0, b1;
            b0.x = bcol0[(size_t)kk * DV];
            b0.y = bcol0[(size_t)(kk + 1) * DV];
            b1.x = bcol1[(size_t)kk * DV];
            b1.y = bcol1[(size_t)(kk + 1) * DV];

            c0 = __builtin_amdgcn_wmma_f32_16x16x4_f32(
                     false, a, false, b0, (short)0, c0, false, false);
            c1 = __builtin_amdgcn_wmma_f32_16x16x4_f32(
                     false, a, false, b1, (short)0, c1, false, false);
        }
    }

    float* orow = out + ((size_t)(b * QLEN + qbase + 8 * half)) * DV + vbase + lrow;
#pragma unroll
    for (int r = 0; r < 8; ++r) {
        orow[(size_t)r * DV]      = c0[r];
        orow[(size_t)r * DV + 16] = c1[r];
    }
}

// ---------------------------------------------------------------------------
extern "C" void kernel_launch(void* const* d_in, const int* in_sizes, int n_in,
                              void* d_out, int out_size, void* d_ws, size_t ws_size,
                              hipStream_t stream) {
    const float* querys     = (const float*)d_in[0];
    const float* keys       = (const float*)d_in[1];
    const float* values     = (const float*)d_in[2];
    const int*   valid_lens = (const int*)d_in[3];
    const float* Wq         = (const float*)d_in[4];
    const float* Wk         = (const float*)d_in[5];
    const float* wv         = (const float*)d_in[6];
    float*       out        = (float*)d_out;

    // workspace layout: qproj (1MB) | kproj (1MB) | scores/attn (16MB)
    float* qproj  = (float*)d_ws;
    float* kproj  = qproj + (size_t)BATCH * QLEN * NH;
    float* scores = kproj + (size_t)BATCH * KLEN * NH;

    // 1) projections: 4096 rows each, 4 rows per block
    proj_kernel<<<dim3((BATCH * QLEN) / 4), 256, 0, stream>>>(querys, Wq, qproj);
    proj_kernel<<<dim3((BATCH * KLEN) / 4), 256, 0, stream>>>(keys,   Wk, kproj);

    // 2) additive scores with tanh (dominant VALU work)
    score_kernel<<<dim3(KLEN / 16, QLEN / 16, BATCH), 256, 0, stream>>>(
        qproj, kproj, wv, scores);

    // 3) masked softmax over K, in place
    softmax_kernel<<<dim3(QLEN, BATCH), 256, 0, stream>>>(scores, valid_lens);

    // 4) attn @ values via f32 WMMA, 16x32 tile per wave
    av_wmma_kernel<<<dim3(DV / 32, QLEN / 64, BATCH), 128, 0, stream>>>(
        scores, values, out);
}